// AttnHGCN_16724602650759
// MI455X (gfx1250) — compile-verified
//
#include <hip/hip_runtime.h>
#include <math.h>

// CDNA5 / gfx1250 implementation of the 2-hop attention GNN.
// Workspace layout (floats):
//   [seg_max: nEnt][seg_sum: nEnt][agg: nEnt*64][s_buf: E][ent1: nEnt*64]
// ~56 MB total. seg_max/seg_sum/agg are contiguous so one zero pass covers them.

typedef __attribute__((ext_vector_type(2))) float v2f;
typedef __attribute__((ext_vector_type(8))) float v8f;

#define CH 64
#define L2EPS 1e-12f

__global__ void zero_kernel(float* __restrict__ p, long n) {
    long i = (long)blockIdx.x * blockDim.x + threadIdx.x;
    long stride = (long)gridDim.x * blockDim.x;
    for (; i < n; i += stride) p[i] = 0.0f;
}

// One wave computes 16 edge scores via V_WMMA_F32_16X16X4_F32 (exact f32 math):
//   A = (h .* r) tile [16 x 4], B = t^T tile [4 x 16], accumulate K=64 in 16 steps.
//   Desired dots are the diagonal of the 16x16 f32 accumulator.
__global__ void score_kernel(const float* __restrict__ ent,
                             const float* __restrict__ rel_emb,
                             const int* __restrict__ head,
                             const int* __restrict__ tail,
                             const int* __restrict__ etype,
                             float* __restrict__ s_buf,
                             float* __restrict__ seg_max,  // int-bits max, init 0
                             int E) {
    int lane = threadIdx.x & 31;
    int wave = threadIdx.x >> 5;
    int edge_base = (blockIdx.x * (blockDim.x >> 5) + wave) << 4;
    if (edge_base >= E) return;           // wave-uniform

    int m = lane & 15;                    // edge within tile
    int half = lane >> 4;                 // K-half select per ISA A/B layout

    if (edge_base + 16 <= E) {            // wave-uniform: full tile, EXEC all 1s
        int e = edge_base + m;
        int hidx = head[e];
        int tidx = tail[e];
        int ridx = etype[e] - 1;
        const float* rowH = ent + (long)hidx * CH;
        const float* rowT = ent + (long)tidx * CH;
        const float* rowR = rel_emb + (long)ridx * CH;

        v8f acc = {};
#pragma unroll
        for (int k = 0; k < 16; ++k) {
            int c0 = 4 * k + 2 * half;    // lane's 2 channels for this K-step
            v2f hv = *(const v2f*)(rowH + c0);
            v2f tv = *(const v2f*)(rowT + c0);
            v2f rv = *(const v2f*)(rowR + c0);
            v2f a;
            a.x = hv.x * rv.x;
            a.y = hv.y * rv.y;
            // D = A(16x4) * B(4x16) + C, f32 exact
            acc = __builtin_amdgcn_wmma_f32_16x16x4_f32(
                false, a, false, tv, (short)0, acc, false, false);
        }

        // Diagonal lives in lanes 0-7 (vgpr = lane, M = lane) and
        // lanes 24-31 (vgpr = lane-24, M = lane-16).
        int r = -1;
        if (lane < 8) r = lane;
        else if (lane >= 24) r = lane - 24;
        if (r >= 0) {
            float d = acc[0];
#pragma unroll
            for (int i = 1; i < 8; ++i) if (r == i) d = acc[i];
            float s = __expf(d);
            int eo = (lane < 8) ? lane : (lane - 16);   // == m for these lanes
            s_buf[edge_base + eo] = s;
            // s > 0 always, so IEEE bit pattern ordering == int ordering
            atomicMax((int*)seg_max + hidx, __float_as_int(s));
        }
    } else {
        // Partial tile (not hit for E % 16 == 0): scalar fallback, no WMMA.
        int e = edge_base + m;
        if (half == 0 && e < E) {
            int hidx = head[e], tidx = tail[e], ridx = etype[e] - 1;
            const float* rowH = ent + (long)hidx * CH;
            const float* rowT = ent + (long)tidx * CH;
            const float* rowR = rel_emb + (long)ridx * CH;
            float d = 0.0f;
            for (int c = 0; c < CH; ++c) d += rowH[c] * rowR[c] * rowT[c];
            float s = __expf(d);
            s_buf[e] = s;
            atomicMax((int*)seg_max + hidx, __float_as_int(s));
        }
    }
}

__global__ void expsum_kernel(const int* __restrict__ head,
                              float* __restrict__ s_buf,
                              const float* __restrict__ seg_max,
                              float* __restrict__ seg_sum,
                              int E) {
    int e = blockIdx.x * blockDim.x + threadIdx.x;
    if (e >= E) return;
    int h = head[e];
    float ex = __expf(s_buf[e] - seg_max[h]);
    s_buf[e] = ex;                          // in-place: e-value replaces s
    atomicAdd(&seg_sum[h], ex);
}

// 64 threads per edge -> coalesced row gather + contiguous f32 atomics into L2.
__global__ void aggregate_kernel(const float* __restrict__ ent,
                                 const int* __restrict__ head,
                                 const int* __restrict__ tail,
                                 const float* __restrict__ s_buf,
                                 const float* __restrict__ seg_sum,
                                 float* __restrict__ agg,
                                 int E) {
    long tid = (long)blockIdx.x * blockDim.x + threadIdx.x;
    long e = tid >> 6;
    int c = (int)(tid & 63);
    if (e >= E) return;
    int h = head[e];
    int t = tail[e];
    float attn = s_buf[e] / seg_sum[h];
    atomicAdd(&agg[(long)h * CH + c], attn * ent[(long)t * CH + c]);
}

// Wave-per-row L2 normalize (wave32 shuffle reduction), float2 per lane.
__global__ void normalize_kernel(const float* __restrict__ src,
                                 float* __restrict__ dst,
                                 int rows) {
    int lane = threadIdx.x & 31;
    int wave = threadIdx.x >> 5;
    int row = blockIdx.x * (blockDim.x >> 5) + wave;
    if (row >= rows) return;
    v2f v = *(const v2f*)(src + (long)row * CH + lane * 2);
    float ss = v.x * v.x + v.y * v.y;
#pragma unroll
    for (int msk = 16; msk >= 1; msk >>= 1) ss += __shfl_xor(ss, msk, 32);
    float inv = 1.0f / fmaxf(sqrtf(ss), L2EPS);
    v2f o;
    o.x = v.x * inv;
    o.y = v.y * inv;
    *(v2f*)(dst + (long)row * CH + lane * 2) = o;
}

__global__ void user_agg_kernel(const float* __restrict__ ent,
                                const int* __restrict__ iu,
                                const int* __restrict__ ii,
                                const float* __restrict__ w,
                                float* __restrict__ uagg,
                                int E) {
    long tid = (long)blockIdx.x * blockDim.x + threadIdx.x;
    long e = tid >> 6;
    int c = (int)(tid & 63);
    if (e >= E) return;
    atomicAdd(&uagg[(long)iu[e] * CH + c], w[e] * ent[(long)ii[e] * CH + c]);
}

extern "C" void kernel_launch(void* const* d_in, const int* in_sizes, int n_in,
                              void* d_out, int out_size, void* d_ws, size_t ws_size,
                              hipStream_t stream) {
    (void)n_in; (void)out_size; (void)ws_size;
    // setup_inputs order: user_emb(unused), item_emb, relation_emb, edge_index,
    //                     edge_type, inter_edge, inter_edge_w
    const float* item_emb = (const float*)d_in[1];
    const float* rel_emb  = (const float*)d_in[2];
    const int*   edge_index = (const int*)d_in[3];
    const int*   edge_type  = (const int*)d_in[4];
    const int*   inter_edge = (const int*)d_in[5];
    const float* inter_w    = (const float*)d_in[6];

    int nUsers = in_sizes[0] / CH;
    int nEnt   = in_sizes[1] / CH;
    int E      = in_sizes[3] / 2;
    int Eui    = in_sizes[5] / 2;

    const int* head = edge_index;
    const int* tail = edge_index + E;
    const int* iu   = inter_edge;
    const int* ii   = inter_edge + Eui;

    float* ws      = (float*)d_ws;
    float* seg_max = ws;                         // nEnt
    float* seg_sum = seg_max + nEnt;             // nEnt
    float* agg     = seg_sum + nEnt;             // nEnt*CH (also reused as user acc)
    float* s_buf   = agg + (long)nEnt * CH;      // E
    float* ent1    = s_buf + E;                  // nEnt*CH

    float* user_out = (float*)d_out;             // [nUsers, 64]
    float* ent_out  = user_out + (long)nUsers * CH;  // [nEnt, 64]

    const int ZB = 256, ZG = 4096;
    long seg_zero_n = (long)nEnt * (CH + 2);     // seg_max|seg_sum|agg contiguous

    const int edgesPerBlock = 8 * 16;            // 8 waves x 16 edges
    int sgrid = (E + edgesPerBlock - 1) / edgesPerBlock;
    long ethreads = (long)E * CH;

    // ---- hop 1: item_emb -> ent1 ----
    zero_kernel<<<ZG, ZB, 0, stream>>>(seg_max, seg_zero_n);
    score_kernel<<<sgrid, 256, 0, stream>>>(item_emb, rel_emb, head, tail,
                                            edge_type, s_buf, seg_max, E);
    expsum_kernel<<<(E + 255) / 256, 256, 0, stream>>>(head, s_buf, seg_max, seg_sum, E);
    aggregate_kernel<<<(int)((ethreads + 255) / 256), 256, 0, stream>>>(
        item_emb, head, tail, s_buf, seg_sum, agg, E);
    normalize_kernel<<<(nEnt + 7) / 8, 256, 0, stream>>>(agg, ent1, nEnt);

    // ---- hop 2: ent1 -> ent_out (written straight into d_out) ----
    zero_kernel<<<ZG, ZB, 0, stream>>>(seg_max, seg_zero_n);
    score_kernel<<<sgrid, 256, 0, stream>>>(ent1, rel_emb, head, tail,
                                            edge_type, s_buf, seg_max, E);
    expsum_kernel<<<(E + 255) / 256, 256, 0, stream>>>(head, s_buf, seg_max, seg_sum, E);
    aggregate_kernel<<<(int)((ethreads + 255) / 256), 256, 0, stream>>>(
        ent1, head, tail, s_buf, seg_sum, agg, E);
    normalize_kernel<<<(nEnt + 7) / 8, 256, 0, stream>>>(agg, ent_out, nEnt);

    // ---- user aggregation (reuse agg as the user accumulator) ----
    zero_kernel<<<ZG, ZB, 0, stream>>>(agg, (long)nUsers * CH);
    long uthreads = (long)Eui * CH;
    user_agg_kernel<<<(int)((uthreads + 255) / 256), 256, 0, stream>>>(
        ent_out, iu, ii, inter_w, agg, Eui);
    normalize_kernel<<<(nUsers + 7) / 8, 256, 0, stream>>>(agg, user_out, nUsers);
}